// QuantumFrontEndQCNN_80736795230975
// MI455X (gfx1250) — compile-verified
//
#include <hip/hip_runtime.h>
#include <math.h>

typedef __attribute__((ext_vector_type(2))) float v2f;
typedef __attribute__((ext_vector_type(8))) float v8f;

#define PI_F 3.14159265358979323846f

// fp32 WMMA: D(16x16) = A(16x4) * B(4x16) + C   -- v_wmma_f32_16x16x4_f32
static __device__ __forceinline__ v8f wmma_f32_k4(v2f a, v2f b, v8f c) {
  return __builtin_amdgcn_wmma_f32_16x16x4_f32(false, a, false, b, (short)0, c, false, false);
}

// ---------------------------------------------------------------------------
// Setup: build fixed circuit unitary U (64x64 complex), Wq = sign ⊗ out_proj,
// and per-image style angles. One block, 64 threads (thread t = basis column).
// ---------------------------------------------------------------------------
__global__ void qcnn_setup_kernel(const float* __restrict__ style,
                                  const float* __restrict__ style_w,
                                  const float* __restrict__ style_b,
                                  const float* __restrict__ qcnn,
                                  const float* __restrict__ meas,
                                  const float* __restrict__ out_w,
                                  float* __restrict__ Ur, float* __restrict__ Ui,
                                  float* __restrict__ Wq, float* __restrict__ sAng)
{
  __shared__ float sre[4096];
  __shared__ float sim[4096];
  const int t = threadIdx.x;  // 0..63, owns column t; no cross-thread mixing -> no barriers

  for (int a = 0; a < 64; ++a) { sre[a*64+t] = (a == t) ? 1.f : 0.f; sim[a*64+t] = 0.f; }

  // HEA layers: per-wire RY, RZ (scalar params) then ring CNOTs
  for (int l = 0; l < 2; ++l) {
    for (int i = 0; i < 6; ++i) {
      const int mask = 1 << (5 - i);
      {
        float th = 0.5f * qcnn[((l*6 + i)*2 + 0)*3 + 0];
        float c = cosf(th), s = sinf(th);
        for (int a = 0; a < 64; ++a) if (!(a & mask)) {
          int a1 = a | mask;
          float r0 = sre[a*64+t],  i0 = sim[a*64+t];
          float r1 = sre[a1*64+t], i1 = sim[a1*64+t];
          sre[a*64+t]  = c*r0 - s*r1;  sim[a*64+t]  = c*i0 - s*i1;
          sre[a1*64+t] = s*r0 + c*r1;  sim[a1*64+t] = s*i0 + c*i1;
        }
      }
      {
        float th = 0.5f * qcnn[((l*6 + i)*2 + 1)*3 + 0];
        float ch = cosf(th), sh = sinf(th);
        for (int a = 0; a < 64; ++a) {
          float re = sre[a*64+t], im = sim[a*64+t];
          if (!(a & mask)) { sre[a*64+t] = re*ch + im*sh; sim[a*64+t] = im*ch - re*sh; }
          else             { sre[a*64+t] = re*ch - im*sh; sim[a*64+t] = im*ch + re*sh; }
        }
      }
    }
    for (int i = 0; i < 6; ++i) {
      const int mc = 1 << (5 - i);
      const int mt = 1 << (5 - ((i + 1) % 6));
      for (int a = 0; a < 64; ++a) if ((a & mc) && !(a & mt)) {
        int a1 = a | mt;
        float r = sre[a*64+t], q = sim[a*64+t];
        sre[a*64+t]  = sre[a1*64+t]; sim[a*64+t]  = sim[a1*64+t];
        sre[a1*64+t] = r;            sim[a1*64+t] = q;
      }
    }
  }
  // trainable measurement basis: U3 per wire
  for (int i = 0; i < 6; ++i) {
    const int mask = 1 << (5 - i);
    float th = 0.5f * meas[i*3+0];
    float ph = meas[i*3+1], la = meas[i*3+2];
    float ct = cosf(th), st = sinf(th);
    float m01r = -cosf(la)*st,    m01i = -sinf(la)*st;
    float m10r =  cosf(ph)*st,    m10i =  sinf(ph)*st;
    float m11r =  cosf(ph+la)*ct, m11i =  sinf(ph+la)*ct;
    for (int a = 0; a < 64; ++a) if (!(a & mask)) {
      int a1 = a | mask;
      float r0 = sre[a*64+t],  i0 = sim[a*64+t];
      float r1 = sre[a1*64+t], i1 = sim[a1*64+t];
      sre[a*64+t]  = ct*r0 + m01r*r1 - m01i*i1;
      sim[a*64+t]  = ct*i0 + m01r*i1 + m01i*r1;
      sre[a1*64+t] = m10r*r0 - m10i*i0 + m11r*r1 - m11i*i1;
      sim[a1*64+t] = m10r*i0 + m10i*r0 + m11r*i1 + m11i*r1;
    }
  }
  for (int a = 0; a < 64; ++a) { Ur[a*64+t] = sre[a*64+t]; Ui[a*64+t] = sim[a*64+t]; }

  // Wq[a][c] = sum_i sign_i(a) * out_proj_w[c][i]   (thread t = row a)
  for (int c = 0; c < 64; ++c) {
    float s = 0.f;
    #pragma unroll
    for (int i = 0; i < 6; ++i)
      s += (((t >> (5 - i)) & 1) ? -1.f : 1.f) * out_w[c*6 + i];
    Wq[t*64 + c] = s;
  }

  // per-image style angles: tanh(style @ style_w.T + b) * pi
  if (t < 48) {
    int img = t / 6, i = t % 6;
    float d = 0.f;
    for (int j = 0; j < 128; ++j) d += style[img*128 + j] * style_w[i*128 + j];
    sAng[img*8 + i] = tanhf(d + style_b[i]) * PI_F;
  }
}

// ---------------------------------------------------------------------------
// Pack [res_w(64x576); data_w(6x576); pad->80] into WMMA B-fragment order:
// Bpack[((n*144+s)*32+lane)*2+v] = Wcat[n*16 + lane%16][4s + (lane/16)*2 + v]
// ---------------------------------------------------------------------------
__global__ void qcnn_pack_kernel(const float* __restrict__ res_w,
                                 const float* __restrict__ data_w,
                                 float* __restrict__ Bpack)
{
  int idx = blockIdx.x * 256 + threadIdx.x;
  if (idx >= 46080) return;
  int v    = idx & 1;
  int lane = (idx >> 1) & 31;
  int s    = (idx >> 6) % 144;
  int n    = idx / 9216;          // 144*64
  int k    = (s << 2) + ((lane >> 4) << 1) + v;
  int row  = (n << 4) + (lane & 15);
  float val = 0.f;
  if (row < 64)      val = res_w[row * 576 + k];
  else if (row < 70) val = data_w[(row - 64) * 576 + k];
  Bpack[idx] = val;
}

// ---------------------------------------------------------------------------
// Main: one WG = one image b, two pixel rows (128 pixels). 8 waves, each owns
// a 16-pixel M-tile. Stage 1: implicit-GEMM unfold conv (K=576, 5 N-tiles of
// 16: 64 res channels + 6 data angles). Stage 2: psi0 build. Stage 3: WMMA
// psi0 x U^T (re,im) -> probs. Stage 4: WMMA probs x Wq + res + biases.
// ---------------------------------------------------------------------------
__global__ __launch_bounds__(256) void qcnn_main_kernel(
    const float* __restrict__ x,
    const float* __restrict__ data_b,
    const float* __restrict__ res_b,
    const float* __restrict__ out_b,
    const float* __restrict__ gUr,
    const float* __restrict__ gUi,
    const float* __restrict__ gWq,
    const float* __restrict__ gStyle,
    const float* __restrict__ gBp,
    float* __restrict__ out)
{
  extern __shared__ float smem[];
  float* xs   = smem;            // [64][4][66] = 16896   (union with Pm/PRm)
  float* Pm   = smem;            // [128][68]  = 8704     psi0
  float* PRm  = smem + 8704;     // [128][68]  = 8704     probs
  float* Ur   = smem + 17408;    // [64][65]
  float* Ui   = Ur + 4160;       // [64][65]
  float* resS = Ui + 4160;       // [128][65]
  float* thS  = resS + 8320;     // [128][8]
  int*   offT = (int*)(thS + 1024);  // [576]

  const int tid = threadIdx.x;
  const int b   = blockIdx.x >> 5;
  const int h0  = (blockIdx.x & 31) << 1;

  // stage U into padded LDS
  for (int i = tid; i < 4096; i += 256) {
    int a = i >> 6, k = i & 63;
    Ur[a*65 + k] = gUr[i];
    Ui[a*65 + k] = gUi[i];
  }
  // patch element k -> LDS offset (row/col within halo tile)
  for (int k = tid; k < 576; k += 256) {
    int c = k / 9, tt = k % 9;
    offT[k] = ((c << 2) + tt / 3) * 66 + (tt % 3);
  }
  // x halo tile: 64 ch x rows h0-1..h0+2 x cols -1..64 (zero padded)
  for (int i = tid; i < 64*4*66; i += 256) {
    int c = i / 264, rr = i % 264, r = rr / 66, col = rr % 66;
    int h = h0 - 1 + r, w = col - 1;
    float v = 0.f;
    if ((unsigned)h < 64u && (unsigned)w < 64u)
      v = x[(size_t)(b*64 + c) * 4096 + h * 64 + w];
    xs[i] = v;
  }
  __syncthreads();

  const int lane = tid & 31;
  const int hf   = lane >> 4;       // A/B fragment half (K phase)
  const int lr   = lane & 15;       // row (A) / col (B,D) within tile
  const int m0   = (tid >> 5) << 4; // wave's pixel tile base
  const int m    = m0 + lr;
  const int baseM = ((m >> 6) * 66) + (m & 63);

  // ---- Stage 1: conv-as-GEMM, K=576 in 144 steps of 4, 5 N-tiles ----
  v8f acc[5];
  const v8f vz = {0.f,0.f,0.f,0.f,0.f,0.f,0.f,0.f};
  #pragma unroll
  for (int n = 0; n < 5; ++n) acc[n] = vz;

  const float2* Bp2 = (const float2*)gBp;
  for (int s = 0; s < 144; ++s) {
    int kk = (s << 2) + (hf << 1);
    v2f A;
    A.x = xs[offT[kk]     + baseM];
    A.y = xs[offT[kk + 1] + baseM];
    int bbase = (s << 5) + lane;
    #pragma unroll
    for (int n = 0; n < 5; ++n) {
      float2 bv = Bp2[n * 4608 + bbase];
      v2f B; B.x = bv.x; B.y = bv.y;
      acc[n] = wmma_f32_k4(A, B, acc[n]);
    }
  }

  // res channels -> LDS (D layout: lane holds col lr, rows r + 8*hf)
  #pragma unroll
  for (int n = 0; n < 4; ++n) {
    int ch = (n << 4) + lr;
    #pragma unroll
    for (int r = 0; r < 8; ++r)
      resS[(m0 + r + (hf << 3)) * 65 + ch] = acc[n][r];
  }
  // encoding angles: theta = tanh(d + b)*pi + style_angle
  if (lr < 6) {
    float db = data_b[lr];
    float sa = gStyle[(b << 3) + lr];
    #pragma unroll
    for (int r = 0; r < 8; ++r) {
      int mm = m0 + r + (hf << 3);
      thS[(mm << 3) + lr] = tanhf(acc[4][r] + db) * PI_F + sa;
    }
  }
  __syncthreads();

  // ---- Stage 2: build product state psi0 per pixel (xs region reused) ----
  if (tid < 128) {
    float cc[6], ss2[6];
    #pragma unroll
    for (int i = 0; i < 6; ++i) {
      float th = thS[(tid << 3) + i] * 0.5f;
      cc[i] = cosf(th); ss2[i] = sinf(th);
    }
    float p[64];
    p[0] = 1.f;
    #pragma unroll
    for (int i = 0; i < 6; ++i) {
      #pragma unroll
      for (int a = (1 << i) - 1; a >= 0; --a) {
        float tv = p[a];
        p[2*a + 1] = tv * ss2[i];
        p[2*a]     = tv * cc[i];
      }
    }
    #pragma unroll
    for (int a = 0; a < 64; a += 4)
      *(float4*)&Pm[tid * 68 + a] = make_float4(p[a], p[a+1], p[a+2], p[a+3]);
  }
  __syncthreads();

  // ---- Stage 3: psi = U * psi0 via WMMA; probs = re^2 + im^2 ----
  #pragma unroll 1
  for (int nt = 0; nt < 4; ++nt) {
    int acol = (nt << 4) + lr;
    v8f ar = vz, ai = vz;
    for (int s = 0; s < 16; ++s) {
      int k = (s << 2) + (hf << 1);
      v2f A;  A.x  = Pm[m * 68 + k];        A.y  = Pm[m * 68 + k + 1];
      v2f Br; Br.x = Ur[acol * 65 + k];     Br.y = Ur[acol * 65 + k + 1];
      v2f Bi; Bi.x = Ui[acol * 65 + k];     Bi.y = Ui[acol * 65 + k + 1];
      ar = wmma_f32_k4(A, Br, ar);
      ai = wmma_f32_k4(A, Bi, ai);
    }
    #pragma unroll
    for (int r = 0; r < 8; ++r) {
      int mm = m0 + r + (hf << 3);
      PRm[mm * 68 + acol] = ar[r]*ar[r] + ai[r]*ai[r];
    }
  }

  // ---- Stage 4: out = res + probs @ Wq + (res_b + out_b) ----
  #pragma unroll 1
  for (int nt = 0; nt < 4; ++nt) {
    int ccol = (nt << 4) + lr;
    v8f q = vz;
    for (int s = 0; s < 16; ++s) {
      int k = (s << 2) + (hf << 1);
      v2f A; A.x = PRm[m * 68 + k];          A.y = PRm[m * 68 + k + 1];
      v2f B; B.x = gWq[(k << 6) + ccol];     B.y = gWq[((k + 1) << 6) + ccol];
      q = wmma_f32_k4(A, B, q);
    }
    float bias = res_b[ccol] + out_b[ccol];
    #pragma unroll
    for (int r = 0; r < 8; ++r) {
      int mm = m0 + r + (hf << 3);
      resS[mm * 65 + ccol] += q[r] + bias;
    }
  }
  __syncthreads();

  // ---- coalesced NCHW writeout ----
  for (int idx = tid; idx < 8192; idx += 256) {
    int c = idx >> 7, mm = idx & 127;
    int h = h0 + (mm >> 6), w = mm & 63;
    out[(size_t)(b*64 + c) * 4096 + h * 64 + w] = resS[mm * 65 + c];
  }
}

// ---------------------------------------------------------------------------
extern "C" void kernel_launch(void* const* d_in, const int* in_sizes, int n_in,
                              void* d_out, int out_size, void* d_ws, size_t ws_size,
                              hipStream_t stream) {
  (void)in_sizes; (void)n_in; (void)out_size; (void)ws_size;
  const float* x       = (const float*)d_in[0];
  const float* style   = (const float*)d_in[1];
  const float* data_w  = (const float*)d_in[2];
  const float* data_b  = (const float*)d_in[3];
  const float* style_w = (const float*)d_in[4];
  const float* style_b = (const float*)d_in[5];
  const float* qcnn    = (const float*)d_in[6];
  const float* meas    = (const float*)d_in[7];
  const float* out_w   = (const float*)d_in[8];
  const float* out_b   = (const float*)d_in[9];
  const float* res_w   = (const float*)d_in[10];
  const float* res_b   = (const float*)d_in[11];

  float* ws   = (float*)d_ws;
  float* Ur   = ws;              // 4096
  float* Ui   = ws + 4096;       // 4096
  float* Wq   = ws + 8192;       // 4096
  float* sAng = ws + 12288;      // 64
  float* Bp   = ws + 12352;      // 46080

  hipLaunchKernelGGL(qcnn_setup_kernel, dim3(1), dim3(64), 0, stream,
                     style, style_w, style_b, qcnn, meas, out_w, Ur, Ui, Wq, sAng);
  hipLaunchKernelGGL(qcnn_pack_kernel, dim3(180), dim3(256), 0, stream,
                     res_w, data_w, Bp);
  hipLaunchKernelGGL(qcnn_main_kernel, dim3(256), dim3(256), 142592 /* dyn LDS bytes */, stream,
                     x, data_b, res_b, out_b, Ur, Ui, Wq, sAng, Bp, (float*)d_out);
}